// MultiHeadAttention_37220186587685
// MI455X (gfx1250) — compile-verified
//
#include <hip/hip_runtime.h>
#include <math.h>

#define D_MODEL 1024
#define N_HEADS 16
#define D_HEAD  64
#define SEQ_T   2048
#define BATCH   2
#define M_TOT   (BATCH * SEQ_T)   // 4096 tokens

typedef float v2f __attribute__((ext_vector_type(2)));
typedef float v8f __attribute__((ext_vector_type(8)));

// Native fp32 WMMA: D(16x16,f32) = A(16x4,f32) * B(4x16,f32) + C
__device__ __forceinline__ v8f wmma_f32(v2f a, v2f b, v8f c) {
  return __builtin_amdgcn_wmma_f32_16x16x4_f32(
      false, a, false, b, (short)0, c, false, false);
}

// ---------------------------------------------------------------------------
// Y[m,n] = sum_k X[m,k] * W[n,k] + bias[n]     (torch y = x @ W^T + b)
// grid = (M/128, N/64), block = 256 threads (8 wave32). K % 16 == 0.
// Block tile 128x64; wave grid 4(M) x 2(N); each wave computes a 2x2 grid of
// 16x16 WMMA tiles. Explicit LDS double-buffering: global loads for panel p+1
// are issued before the WMMAs of panel p, so HBM latency hides behind compute.
// One barrier per panel is sufficient (compute < store < barrier per wave).
// LDS strides padded to 17 (gcd(17,64)==1 -> conflict-free transposed reads).
// ---------------------------------------------------------------------------
__global__ __launch_bounds__(256)
void gemm_xwT_bias(const float* __restrict__ X, const float* __restrict__ W,
                   const float* __restrict__ bias, float* __restrict__ Y,
                   int M, int N, int Kd) {
  __shared__ float Xs[2][128][17];
  __shared__ float Ws[2][64][17];

  const int tid = threadIdx.x;
  const int wv  = tid >> 5;          // wave id 0..7
  const int wm  = wv & 3;            // wave row 0..3  (32 M-rows each)
  const int wn  = wv >> 2;           // wave col 0..1  (32 N-cols each)
  const int ln  = tid & 15;          // lane % 16
  const int g   = (tid >> 4) & 1;    // lane / 16 within wave
  const int m0  = blockIdx.x * 128;
  const int n0  = blockIdx.y * 64;

  // Per-thread staging slots for the cooperative tile loads.
  const int rX = tid >> 1, cX = (tid & 1) * 8;   // X tile 128x16: 8 f/thread
  const int rW = tid >> 2, cW = (tid & 3) * 4;   // W tile  64x16: 4 f/thread
  const float* xsrc = X + (size_t)(m0 + rX) * Kd + cX;
  const float* wsrc = W + (size_t)(n0 + rW) * Kd + cW;

  v8f acc00 = {0.f,0.f,0.f,0.f,0.f,0.f,0.f,0.f};
  v8f acc01 = acc00, acc10 = acc00, acc11 = acc00;

  // --- prologue: stage panel 0 into buffer 0 ---
  {
    float4 x0 = *(const float4*)(xsrc);
    float4 x1 = *(const float4*)(xsrc + 4);
    float4 w0 = *(const float4*)(wsrc);
    Xs[0][rX][cX+0]=x0.x; Xs[0][rX][cX+1]=x0.y;
    Xs[0][rX][cX+2]=x0.z; Xs[0][rX][cX+3]=x0.w;
    Xs[0][rX][cX+4]=x1.x; Xs[0][rX][cX+5]=x1.y;
    Xs[0][rX][cX+6]=x1.z; Xs[0][rX][cX+7]=x1.w;
    Ws[0][rW][cW+0]=w0.x; Ws[0][rW][cW+1]=w0.y;
    Ws[0][rW][cW+2]=w0.z; Ws[0][rW][cW+3]=w0.w;
  }
  __syncthreads();

  const int npan = Kd >> 4;
  for (int p = 0; p < npan; ++p) {
    const int cur = p & 1;
    const bool have_next = (p + 1 < npan);

    // Issue next panel's global loads now; no wait until the ds_store below,
    // so they overlap the 16 WMMAs of the current panel.
    float4 nx0, nx1, nw0;
    if (have_next) {
      nx0 = *(const float4*)(xsrc + (p + 1) * 16);
      nx1 = *(const float4*)(xsrc + (p + 1) * 16 + 4);
      nw0 = *(const float4*)(wsrc + (p + 1) * 16);
    }

    // 16 WMMAs over the 16-deep panel (4 per K-step of 4).
#pragma unroll
    for (int kk = 0; kk < 4; ++kk) {
      const int kb = kk * 4 + 2 * g;
      v2f a0, a1, b0, b1;
      a0.x = Xs[cur][wm*32 +      ln][kb]; a0.y = Xs[cur][wm*32 +      ln][kb+1];
      a1.x = Xs[cur][wm*32 + 16 + ln][kb]; a1.y = Xs[cur][wm*32 + 16 + ln][kb+1];
      b0.x = Ws[cur][wn*32 +      ln][kb]; b0.y = Ws[cur][wn*32 +      ln][kb+1];
      b1.x = Ws[cur][wn*32 + 16 + ln][kb]; b1.y = Ws[cur][wn*32 + 16 + ln][kb+1];
      acc00 = wmma_f32(a0, b0, acc00);
      acc01 = wmma_f32(a0, b1, acc01);
      acc10 = wmma_f32(a1, b0, acc10);
      acc11 = wmma_f32(a1, b1, acc11);
    }

    if (have_next) {
      const int nxt = cur ^ 1;
      Xs[nxt][rX][cX+0]=nx0.x; Xs[nxt][rX][cX+1]=nx0.y;
      Xs[nxt][rX][cX+2]=nx0.z; Xs[nxt][rX][cX+3]=nx0.w;
      Xs[nxt][rX][cX+4]=nx1.x; Xs[nxt][rX][cX+5]=nx1.y;
      Xs[nxt][rX][cX+6]=nx1.z; Xs[nxt][rX][cX+7]=nx1.w;
      Ws[nxt][rW][cW+0]=nw0.x; Ws[nxt][rW][cW+1]=nw0.y;
      Ws[nxt][rW][cW+2]=nw0.z; Ws[nxt][rW][cW+3]=nw0.w;
    }
    __syncthreads();
  }

  // Epilogue: bias + store. C layout: row = v + 8g, col = ln.
  const float bv0 = bias[n0 + wn * 32 +      ln];
  const float bv1 = bias[n0 + wn * 32 + 16 + ln];
  const int mr0 = m0 + wm * 32 +      8 * g;
  const int mr1 = m0 + wm * 32 + 16 + 8 * g;
  const int nc0 = n0 + wn * 32 +      ln;
  const int nc1 = n0 + wn * 32 + 16 + ln;
#pragma unroll
  for (int v = 0; v < 8; ++v) {
    Y[(size_t)(mr0 + v) * N + nc0] = acc00[v] + bv0;
    Y[(size_t)(mr0 + v) * N + nc1] = acc01[v] + bv1;
    Y[(size_t)(mr1 + v) * N + nc0] = acc10[v] + bv0;
    Y[(size_t)(mr1 + v) * N + nc1] = acc11[v] + bv1;
  }
}

// ---------------------------------------------------------------------------
// Causal flash-attention per (b,h).
// grid = (T/64, B*H), block = 128 threads (4 wave32); wave w owns q-rows
// [q0+16w, +16). Streams 16-key blocks with double-buffered K/V LDS tiles:
// next block's global loads are in flight during the current block's
// S = Q K^T WMMAs, online softmax, and O += P V WMMAs.
// ---------------------------------------------------------------------------
__global__ __launch_bounds__(128)
void attn_flash(const float* __restrict__ Q, const float* __restrict__ K,
                const float* __restrict__ V, float* __restrict__ O) {
  __shared__ float Kt[2][16][68];     // stride 68 -> conflict-free column reads
  __shared__ float Vt[2][16][68];
  __shared__ float Ps[4][16][17];     // per-wave P bounce (C-layout -> A-layout)

  const int tid = threadIdx.x;
  const int wv  = tid >> 5;
  const int ln  = tid & 15;
  const int g   = (tid >> 4) & 1;
  const int bh  = blockIdx.y;
  const int b   = bh / N_HEADS;
  const int h   = bh % N_HEADS;
  const int q0  = blockIdx.x * 64;

  const size_t base = (size_t)b * SEQ_T * D_MODEL + (size_t)h * D_HEAD;
  const float NEG_INF = -__builtin_inff();

  // --- preload Q A-fragments: 16 x 64, 16 K-steps of 4 ---
  v2f qf[16];
  {
    const float* qp = Q + base + (size_t)(q0 + wv * 16 + ln) * D_MODEL;
#pragma unroll
    for (int kk = 0; kk < 16; ++kk) {
      const int d = kk * 4 + 2 * g;
      qf[kk].x = qp[d];
      qf[kk].y = qp[d + 1];
    }
  }

  v8f o0 = {0.f,0.f,0.f,0.f,0.f,0.f,0.f,0.f};
  v8f o1 = o0, o2 = o0, o3 = o0;
  float m_i[8], l_i[8];
#pragma unroll
  for (int v = 0; v < 8; ++v) { m_i[v] = NEG_INF; l_i[v] = 0.f; }

  // Cooperative tile-load slots: 16 keys x 64 dims, 8 floats/thread per array.
  const int rT = tid >> 3;
  const int cT = (tid & 7) * 8;
  const float* kp0 = K + base + (size_t)rT * D_MODEL + cT;
  const float* vp0 = V + base + (size_t)rT * D_MODEL + cT;

  const int nkb = q0 / 16 + 4;        // key blocks with k0 <= q0+63 (causal)

  // --- prologue: stage key-block 0 into buffer 0 ---
  {
    float4 ka = ((const float4*)kp0)[0], kb4 = ((const float4*)kp0)[1];
    float4 va = ((const float4*)vp0)[0], vb4 = ((const float4*)vp0)[1];
    Kt[0][rT][cT+0]=ka.x;  Kt[0][rT][cT+1]=ka.y;
    Kt[0][rT][cT+2]=ka.z;  Kt[0][rT][cT+3]=ka.w;
    Kt[0][rT][cT+4]=kb4.x; Kt[0][rT][cT+5]=kb4.y;
    Kt[0][rT][cT+6]=kb4.z; Kt[0][rT][cT+7]=kb4.w;
    Vt[0][rT][cT+0]=va.x;  Vt[0][rT][cT+1]=va.y;
    Vt[0][rT][cT+2]=va.z;  Vt[0][rT][cT+3]=va.w;
    Vt[0][rT][cT+4]=vb4.x; Vt[0][rT][cT+5]=vb4.y;
    Vt[0][rT][cT+6]=vb4.z; Vt[0][rT][cT+7]=vb4.w;
  }
  __syncthreads();

  for (int kb = 0; kb < nkb; ++kb) {
    const int cur = kb & 1;
    const bool have_next = (kb + 1 < nkb);

    // Issue next key-block loads now; waited on only at the ds_store below.
    float4 nka, nkb4, nva, nvb4;
    if (have_next) {
      const float* kp = kp0 + (size_t)(kb + 1) * 16 * D_MODEL;
      const float* vp = vp0 + (size_t)(kb + 1) * 16 * D_MODEL;
      nka  = ((const float4*)kp)[0]; nkb4 = ((const float4*)kp)[1];
      nva  = ((const float4*)vp)[0]; nvb4 = ((const float4*)vp)[1];
    }

    // --- S = (Q K^T) tile, 16 q-rows x 16 keys, reduce over d=64 ---
    v8f s = {0.f,0.f,0.f,0.f,0.f,0.f,0.f,0.f};
#pragma unroll
    for (int kk = 0; kk < 16; ++kk) {
      const int d = kk * 4 + 2 * g;
      v2f bf;                         // B[d][key] = K[key][d]
      bf.x = Kt[cur][ln][d];
      bf.y = Kt[cur][ln][d + 1];
      s = wmma_f32(qf[kk], bf, s);
    }

    // --- online softmax; C layout: row = v+8g (q), col = ln (key) ---
#pragma unroll
    for (int v = 0; v < 8; ++v) {
      float sv = s[v] * 0.125f;                   // 1/sqrt(64)
      const int qg = q0 + wv * 16 + v + 8 * g;
      const int kg = kb * 16 + ln;
      if (kg > qg) sv = NEG_INF;                  // causal mask

      float rm = sv;                              // row max over 16 lanes
      rm = fmaxf(rm, __shfl_xor(rm, 8, 16));
      rm = fmaxf(rm, __shfl_xor(rm, 4, 16));
      rm = fmaxf(rm, __shfl_xor(rm, 2, 16));
      rm = fmaxf(rm, __shfl_xor(rm, 1, 16));

      const float mn = fmaxf(m_i[v], rm);
      float corr, p;
      if (mn == NEG_INF) { corr = 1.0f; p = 0.0f; }   // fully masked block
      else { corr = __expf(m_i[v] - mn); p = __expf(sv - mn); }
      m_i[v] = mn;

      float ps = p;                               // row sum over 16 lanes
      ps += __shfl_xor(ps, 8, 16);
      ps += __shfl_xor(ps, 4, 16);
      ps += __shfl_xor(ps, 2, 16);
      ps += __shfl_xor(ps, 1, 16);
      l_i[v] = l_i[v] * corr + ps;

      o0[v] *= corr; o1[v] *= corr; o2[v] *= corr; o3[v] *= corr;
      Ps[wv][v + 8 * g][ln] = p;                  // spill P in C layout
    }

    // --- O += P V : re-read P as A-frags (same-wave DS is in-order) ---
#pragma unroll
    for (int kk = 0; kk < 4; ++kk) {
      const int kr = kk * 4 + 2 * g;
      v2f a;
      a.x = Ps[wv][ln][kr];
      a.y = Ps[wv][ln][kr + 1];
      v2f b0, b1, b2, b3;                         // b[v] = V[key][d]
      b0.x = Vt[cur][kr][ 0 + ln]; b0.y = Vt[cur][kr + 1][ 0 + ln];
      b1.x = Vt[cur][kr][16 + ln]; b1.y = Vt[cur][kr + 1][16 + ln];
      b2.x = Vt[cur][kr][32 + ln]; b2.y = Vt[cur][kr + 1][32 + ln];
      b3.x = Vt[cur][kr][48 + ln]; b3.y = Vt[cur][kr + 1][48 + ln];
      o0 = wmma_f32(a, b0, o0);
      o1 = wmma_f32(a, b1, o1);
      o2 = wmma_f32(a, b2, o2);
      o3 = wmma_f32(a, b3, o3);
    }

    if (have_next) {
      const int nxt = cur ^ 1;
      Kt[nxt][rT][cT+0]=nka.x;  Kt[nxt][rT][cT+1]=nka.y;
      Kt[nxt][rT][cT+2]=nka.z;  Kt[nxt][rT][cT+3]=nka.w;
      Kt[nxt][rT][cT+4]=nkb4.x; Kt[nxt][rT][cT+5]=nkb4.y;
      Kt[nxt][rT][cT+6]=nkb4.z; Kt[nxt][rT][cT+7]=nkb4.w;
      Vt[nxt][rT][cT+0]=nva.x;  Vt[nxt][rT][cT+1]=nva.y;
      Vt[nxt][rT][cT+2]=nva.z;  Vt[nxt][rT][cT+3]=nva.w;
      Vt[nxt][rT][cT+4]=nvb4.x; Vt[nxt][rT][cT+5]=nvb4.y;
      Vt[nxt][rT][cT+6]=nvb4.z; Vt[nxt][rT][cT+7]=nvb4.w;
    }
    __syncthreads();
  }

  // --- normalize and store attention output, layout [B, T, H*D] ---
#pragma unroll
  for (int v = 0; v < 8; ++v) {
    const float inv = (l_i[v] > 0.f) ? 1.0f / l_i[v] : 0.f;
    const int t = q0 + wv * 16 + v + 8 * g;
    float* op = O + base + (size_t)t * D_MODEL;
    op[ 0 + ln] = o0[v] * inv;
    op[16 + ln] = o1[v] * inv;
    op[32 + ln] = o2[v] * inv;
    op[48 + ln] = o3[v] * inv;
  }
}

// ---------------------------------------------------------------------------
extern "C" void kernel_launch(void* const* d_in, const int* in_sizes, int n_in,
                              void* d_out, int out_size, void* d_ws, size_t ws_size,
                              hipStream_t stream) {
  const float* x  = (const float*)d_in[0];
  const float* Wq = (const float*)d_in[1];
  const float* bq = (const float*)d_in[2];
  const float* Wk = (const float*)d_in[3];
  const float* bk = (const float*)d_in[4];
  const float* Wv = (const float*)d_in[5];
  const float* bv = (const float*)d_in[6];
  const float* Wo = (const float*)d_in[7];
  const float* bo = (const float*)d_in[8];

  const size_t mat = (size_t)M_TOT * D_MODEL;   // 4096*1024 floats = 16 MB
  float* Qb = (float*)d_ws;                     // workspace: 4 * 16 MB
  float* Kb = Qb + mat;
  float* Vb = Kb + mat;
  float* Ab = Vb + mat;

  const dim3 ggrid(M_TOT / 128, D_MODEL / 64);  // (32, 16)
  gemm_xwT_bias<<<ggrid, 256, 0, stream>>>(x,  Wq, bq, Qb, M_TOT, D_MODEL, D_MODEL);
  gemm_xwT_bias<<<ggrid, 256, 0, stream>>>(x,  Wk, bk, Kb, M_TOT, D_MODEL, D_MODEL);
  gemm_xwT_bias<<<ggrid, 256, 0, stream>>>(x,  Wv, bv, Vb, M_TOT, D_MODEL, D_MODEL);

  attn_flash<<<dim3(SEQ_T / 64, BATCH * N_HEADS), 128, 0, stream>>>(Qb, Kb, Vb, Ab);

  gemm_xwT_bias<<<ggrid, 256, 0, stream>>>(Ab, Wo, bo, (float*)d_out,
                                           M_TOT, D_MODEL, D_MODEL);
}